// GPFPlus_52999896433002
// MI455X (gfx1250) — compile-verified
//
#include <hip/hip_runtime.h>

typedef float v2f __attribute__((ext_vector_type(2)));
typedef float v8f __attribute__((ext_vector_type(8)));

#define TOKEN_DIM 128
#define TOKEN_NUM 16
#define WAVES_PER_BLOCK 8
#define BLOCK_THREADS (WAVES_PER_BLOCK * 32)

// One wave32 processes a 16-node tile.
// GEMM1 (transposed): S^T[t,m] = sum_k T[t,k] * X[m,k]   (32 x wmma 16x16x4 f32)
// softmax over t per node m (8 in-lane + one shfl_xor(16))
// GEMM2 (transposed): P^T[d,m] = sum_t T[t,d] * att[m,t] (32 x wmma 16x16x4 f32)
// out[m,d] = x[m,d] + P^T[d,m]
__global__ __launch_bounds__(BLOCK_THREADS) void gpfplus_prompt_kernel(
    const float* __restrict__ x,
    const float* __restrict__ te,
    float* __restrict__ out,
    int n_nodes)
{
    __shared__ __align__(16) float t_lds[TOKEN_NUM * TOKEN_DIM]; // 8 KB

    const int tid = threadIdx.x;
    // Stage token_embeds (16x128 f32) into LDS once per block.
    for (int i = tid; i < TOKEN_NUM * TOKEN_DIM; i += BLOCK_THREADS)
        t_lds[i] = te[i];
    __syncthreads();

    const int lane = tid & 31;
    const int wave = tid >> 5;
    const int tile = blockIdx.x * WAVES_PER_BLOCK + wave;
    const int node_base = tile * 16;
    if (node_base + 16 > n_nodes) return;   // wave-uniform exit (EXEC stays all-ones)

    const int m = lane & 15;  // column index: node (B/C roles) or M-row (A role)
    const int h = lane >> 4;  // half-wave select

    // ---------------- GEMM1: S^T = T @ X^T, K=128 in steps of 4 ----------------
    // A 16x4 f32 layout: lane (h*16+t): V0=A[t,2h], V1=A[t,2h+1]
    // B 4x16 f32 layout: lane (h*16+m): V0=B[2h,m], V1=B[2h+1,m]
    v8f acc = {};
    const float* xrow = x + (size_t)(node_base + m) * TOKEN_DIM + 2 * h;
    const float* trow = t_lds + m * TOKEN_DIM + 2 * h;
#pragma unroll 8
    for (int kb = 0; kb < TOKEN_DIM; kb += 4) {
        v2f a = *(const v2f*)(trow + kb);   // ds_load_b64: T[token=m][kb+2h .. +1]
        v2f b = *(const v2f*)(xrow + kb);   // global_load_b64: X[node=m][kb+2h .. +1]
        acc = __builtin_amdgcn_wmma_f32_16x16x4_f32(
            false, a, false, b, (short)0, acc, false, false);
    }
    // acc[r] = S^T[8h + r, node m]  -> lane holds 8 of the 16 token scores of node m,
    // the other 8 live in lane ^ 16.

    // ---------------- softmax over the 16 tokens of node m ----------------
    float mx = acc[0];
#pragma unroll
    for (int r = 1; r < 8; ++r) mx = fmaxf(mx, acc[r]);
    mx = fmaxf(mx, __shfl_xor(mx, 16, 32));

    float e[8];
    float sum = 0.0f;
#pragma unroll
    for (int r = 0; r < 8; ++r) {
        e[r] = __expf(acc[r] - mx);
        sum += e[r];
    }
    sum += __shfl_xor(sum, 16, 32);
    const float inv = 1.0f / sum;

    float al[8];                 // att for tokens 8h .. 8h+7 of node m
#pragma unroll
    for (int r = 0; r < 8; ++r) al[r] = e[r] * inv;

    // Broadcast both halves so every lane holds all 16 att values of its node.
    float af[16];
#pragma unroll
    for (int j = 0; j < 8; ++j) {
        float o = __shfl_xor(al[j], 16, 32);
        af[j]     = h ? o     : al[j];
        af[8 + j] = h ? al[j] : o;
    }

    // ---------------- GEMM2: P^T = T^T @ att^T, 8 dim-chunks, K=16 ----------------
#pragma unroll
    for (int dc = 0; dc < 8; ++dc) {
        v8f p = {};
        const int dbase = dc * 16 + m;      // here m plays the A-matrix M-row (dim)
#pragma unroll
        for (int tb = 0; tb < TOKEN_NUM; tb += 4) {
            // A[d, k] = T[tb + 2h + k', dc*16 + d]
            v2f a;
            a.x = t_lds[(tb + 2 * h + 0) * TOKEN_DIM + dbase];
            a.y = t_lds[(tb + 2 * h + 1) * TOKEN_DIM + dbase];
            // B[k, m] = att[m, tb + 2h + k']  (compile-time tb -> pure v_cndmask)
            v2f b;
            b.x = h ? af[tb + 2] : af[tb + 0];
            b.y = h ? af[tb + 3] : af[tb + 1];
            p = __builtin_amdgcn_wmma_f32_16x16x4_f32(
                false, a, false, b, (short)0, p, false, false);
        }
        // p[r] = prompt[node m, dc*16 + 8h + r] ; residual add + store (32B/lane)
        const float* xr = x  + (size_t)(node_base + m) * TOKEN_DIM + dc * 16 + 8 * h;
        float*      orw = out + (size_t)(node_base + m) * TOKEN_DIM + dc * 16 + 8 * h;
        float4 x0 = *(const float4*)(xr);
        float4 x1 = *(const float4*)(xr + 4);
        float4 r0 = make_float4(p[0] + x0.x, p[1] + x0.y, p[2] + x0.z, p[3] + x0.w);
        float4 r1 = make_float4(p[4] + x1.x, p[5] + x1.y, p[6] + x1.z, p[7] + x1.w);
        *(float4*)(orw)     = r0;
        *(float4*)(orw + 4) = r1;
    }
}

extern "C" void kernel_launch(void* const* d_in, const int* in_sizes, int n_in,
                              void* d_out, int out_size, void* d_ws, size_t ws_size,
                              hipStream_t stream) {
    const float* x  = (const float*)d_in[0];
    const float* te = (const float*)d_in[1];
    float* out = (float*)d_out;

    const int n_nodes = in_sizes[0] / TOKEN_DIM;      // 400000
    const int tiles   = n_nodes / 16;                 // 25000 (N divisible by 16)
    const int blocks  = (tiles + WAVES_PER_BLOCK - 1) / WAVES_PER_BLOCK;

    gpfplus_prompt_kernel<<<blocks, BLOCK_THREADS, 0, stream>>>(x, te, out, n_nodes);
}